// CosineLoss_90417651516231
// MI455X (gfx1250) — compile-verified
//
#include <hip/hip_runtime.h>

typedef __attribute__((ext_vector_type(2))) float v2f;
typedef __attribute__((ext_vector_type(8))) float v8f;

#define DDIM 512
#define NROWS 4096
#define TILE_M 16
#define WAVES_PER_BLOCK 8
#define THREADS (WAVES_PER_BLOCK * 32)
#define N_TILES (NROWS / TILE_M)               // 256
#define N_BLOCKS (N_TILES / WAVES_PER_BLOCK)   // 32
#define WS_FLOATS (2 * DDIM + 1)               // sx[512], sy[512], diag[1]

// ---------------- kernel 0: zero the workspace accumulators ----------------
__global__ __launch_bounds__(256) void cl_zero_ws(float* __restrict__ ws) {
    int i = blockIdx.x * blockDim.x + threadIdx.x;
    if (i < WS_FLOATS) ws[i] = 0.0f;
}

// ---------------- kernel 1: per-row stats via WMMA ones-trick ----------------
// Each wave owns one 16-row tile. Phase 1: chained V_WMMA_F32_16X16X4_F32 with a
// ones B-matrix reduces sum(x^2), sum(y^2), sum(x*y) per row into the f32
// accumulator (every lane holds 8 row-sums). Phase 2: coalesced reread of the
// tile, scaled by 1/max(norm,eps), accumulated into block LDS then global ws.
__global__ __launch_bounds__(THREADS) void cl_stats(const float* __restrict__ X,
                                                    const float* __restrict__ Y,
                                                    float* __restrict__ ws) {
    __shared__ float s_acc[2 * DDIM];
    __shared__ float s_diag;

    const int tid  = threadIdx.x;
    const int lane = tid & 31;
    const int wave = tid >> 5;

    for (int i = tid; i < 2 * DDIM; i += THREADS) s_acc[i] = 0.0f;
    if (tid == 0) s_diag = 0.0f;
    __syncthreads();

    const int tile = blockIdx.x * WAVES_PER_BLOCK + wave;   // 0..255
    const int row0 = tile * TILE_M;
    const int mrow = lane & 15;
    const int hi   = lane >> 4;                              // 0: K={0,1}, 1: K={2,3}

    // Per-lane base pointer matching the f32 16x4 A-operand layout.
    const float* px = X + (size_t)(row0 + mrow) * DDIM + 2 * hi;
    const float* py = Y + (size_t)(row0 + mrow) * DDIM + 2 * hi;

    v8f cxx = {}, cyy = {}, cxy = {};
    const v2f bones = {1.0f, 1.0f};   // ones B-matrix: D[m][n] = rowsum(A[m][:])

    #pragma unroll 8
    for (int k0 = 0; k0 < DDIM; k0 += 4) {
        v2f ax = *(const v2f*)(px + k0);
        v2f ay = *(const v2f*)(py + k0);
        v2f axx = ax * ax;
        v2f ayy = ay * ay;
        v2f axy = ax * ay;
        cxx = __builtin_amdgcn_wmma_f32_16x16x4_f32(false, axx, false, bones,
                                                    (short)0, cxx, false, false);
        cyy = __builtin_amdgcn_wmma_f32_16x16x4_f32(false, ayy, false, bones,
                                                    (short)0, cyy, false, false);
        cxy = __builtin_amdgcn_wmma_f32_16x16x4_f32(false, axy, false, bones,
                                                    (short)0, cxy, false, false);
    }

    // Lanes 0-15 hold rows 0..7 in their 8 accumulator VGPRs, lanes 16-31 rows 8..15.
    float invx[8], invy[8];
    float dpart = 0.0f;
    #pragma unroll
    for (int v = 0; v < 8; ++v) {
        float nx = fmaxf(sqrtf(cxx[v]), 1e-8f);
        float ny = fmaxf(sqrtf(cyy[v]), 1e-8f);
        invx[v] = 1.0f / nx;
        invy[v] = 1.0f / ny;
        dpart = fmaf(cxy[v] * invx[v], invy[v], dpart);   // diag: x_n.y_n/(|x||y|)
    }
    if (lane == 0 || lane == 16) atomicAdd(&s_diag, dpart);

    // Phase 2: accumulate normalized column sums. Lane owns d = 4*(lane+32j)+c.
    float accx[16], accy[16];
    #pragma unroll
    for (int i = 0; i < 16; ++i) { accx[i] = 0.0f; accy[i] = 0.0f; }

    #pragma unroll
    for (int m = 0; m < TILE_M; ++m) {
        const float wx = __shfl(invx[m & 7], (m < 8) ? 0 : 16);
        const float wy = __shfl(invy[m & 7], (m < 8) ? 0 : 16);
        const float4* rx = (const float4*)(X + (size_t)(row0 + m) * DDIM);
        const float4* ry = (const float4*)(Y + (size_t)(row0 + m) * DDIM);
        #pragma unroll
        for (int j = 0; j < 4; ++j) {
            const float4 vx = rx[lane + 32 * j];
            const float4 vy = ry[lane + 32 * j];
            accx[4 * j + 0] = fmaf(vx.x, wx, accx[4 * j + 0]);
            accx[4 * j + 1] = fmaf(vx.y, wx, accx[4 * j + 1]);
            accx[4 * j + 2] = fmaf(vx.z, wx, accx[4 * j + 2]);
            accx[4 * j + 3] = fmaf(vx.w, wx, accx[4 * j + 3]);
            accy[4 * j + 0] = fmaf(vy.x, wy, accy[4 * j + 0]);
            accy[4 * j + 1] = fmaf(vy.y, wy, accy[4 * j + 1]);
            accy[4 * j + 2] = fmaf(vy.z, wy, accy[4 * j + 2]);
            accy[4 * j + 3] = fmaf(vy.w, wy, accy[4 * j + 3]);
        }
    }

    // Combine the block's 8 waves in LDS (ds_add_f32), then one global atomic pass.
    #pragma unroll
    for (int j = 0; j < 4; ++j) {
        const int d0 = 4 * (lane + 32 * j);
        #pragma unroll
        for (int c = 0; c < 4; ++c) {
            atomicAdd(&s_acc[d0 + c],        accx[4 * j + c]);
            atomicAdd(&s_acc[DDIM + d0 + c], accy[4 * j + c]);
        }
    }
    __syncthreads();

    for (int i = tid; i < 2 * DDIM; i += THREADS) atomicAdd(&ws[i], s_acc[i]);
    if (tid == 0) atomicAdd(&ws[2 * DDIM], s_diag);
}

// ---------------- kernel 2: finalize: (sx.sy - diag) / N^2 ----------------
__global__ __launch_bounds__(256) void cl_finalize(const float* __restrict__ ws,
                                                   float* __restrict__ out) {
    __shared__ float red[256];
    const int tid = threadIdx.x;
    float p = ws[tid] * ws[DDIM + tid] + ws[tid + 256] * ws[DDIM + tid + 256];
    red[tid] = p;
    __syncthreads();
    for (int s = 128; s > 0; s >>= 1) {
        if (tid < s) red[tid] += red[tid + s];
        __syncthreads();
    }
    if (tid == 0) {
        const float n2 = (float)NROWS * (float)NROWS;
        out[0] = (red[0] - ws[2 * DDIM]) / n2;
    }
}

extern "C" void kernel_launch(void* const* d_in, const int* in_sizes, int n_in,
                              void* d_out, int out_size, void* d_ws, size_t ws_size,
                              hipStream_t stream) {
    const float* X = (const float*)d_in[0];
    const float* Y = (const float*)d_in[1];
    float* ws  = (float*)d_ws;
    float* out = (float*)d_out;

    cl_zero_ws<<<dim3((WS_FLOATS + 255) / 256), dim3(256), 0, stream>>>(ws);
    cl_stats<<<dim3(N_BLOCKS), dim3(THREADS), 0, stream>>>(X, Y, ws);
    cl_finalize<<<dim3(1), dim3(256), 0, stream>>>(ws, out);
}